// BrainGCN_11390253269178
// MI455X (gfx1250) — compile-verified
//
#include <hip/hip_runtime.h>
#include <hip/hip_bf16.h>

// ---------------------------------------------------------------------------
// BrainGCN on MI455X (gfx1250, wave32).
//   h1 = tanh(Â (x@W1) + b1)   Â = D^-1/2 (A+I) D^-1/2
//   h2 = tanh(Â (h1@W2) + b2)
//   h3 = tanh(h2@Wf1 + bf1);  out = h3@Wf2 + bf2
// Dense transforms use V_WMMA_F32_16X16X4_F32 (fp32 path; GEMMs are only
// ~2.6 GFLOP, the op is scatter-bound, so fp32 WMMA keeps reference precision
// at zero cost). Scatter uses float4 loads + global_atomic_add_f32 (L2-resident:
// whole working set < 192MB L2).
// ---------------------------------------------------------------------------

typedef __attribute__((ext_vector_type(2))) float v2f;
typedef __attribute__((ext_vector_type(8))) float v8f;

#define DEV_INLINE __device__ __forceinline__

// ------------------------- degree / normalization --------------------------

__global__ void deg_init_kernel(float* deg, int n) {
    int i = blockIdx.x * blockDim.x + threadIdx.x;
    if (i < n) deg[i] = 1.0f;  // self-loop contributes 1
}

__global__ void deg_edges_kernel(const long long* __restrict__ dst, float* deg, int e) {
    int i = blockIdx.x * blockDim.x + threadIdx.x;
    if (i < e) atomicAdd(&deg[(int)dst[i]], 1.0f);
}

__global__ void dinv_kernel(float* deg, int n) {
    int i = blockIdx.x * blockDim.x + threadIdx.x;
    if (i < n) deg[i] = __frsqrt_rn(deg[i]);  // deg >= 1 always
}

// ------------------------- WMMA fp32 transform -----------------------------
// T[n x OUT] = X[n x 128] @ W[128 x OUT]
// 8 waves / block. COLT = OUT/16 column tiles, ROWT = 8/COLT row tiles.
// K staged through LDS in chunks of 64 to stay well under the LDS limit.
//
// f32 WMMA 16x16x4 VGPR layout (ISA 7.12.2):
//   A (16x4):  lane m=lane&15 is row M; hi=lane>>4 selects K pair:
//              a.x = A[M][k0+2*hi], a.y = A[M][k0+2*hi+1]
//   B (4x16):  mirrored: b.x = B[k0+2*hi][N=lane&15], b.y = B[k0+2*hi+1][N]
//   C/D (16x16): VGPR r holds M = r + 8*hi, N = lane&15
template <int OUT>
__global__ void __launch_bounds__(256) gcn_transform_wmma(
    const float* __restrict__ X, const float* __restrict__ W,
    float* __restrict__ T, int n) {
    constexpr int K    = 128;
    constexpr int COLT = OUT / 16;   // 8 (OUT=128) or 4 (OUT=64)
    constexpr int ROWT = 8 / COLT;   // 1 or 2
    constexpr int ROWS = ROWT * 16;  // 16 or 32
    constexpr int KC   = 64;         // K chunk staged in LDS

    __shared__ float sW[KC * OUT];   // 32KB (OUT=128) / 16KB (OUT=64)
    __shared__ float sX[ROWS * KC];  // 4KB / 8KB

    const int tid  = threadIdx.x;
    const int wave = tid >> 5;
    const int lane = tid & 31;
    const int rt   = wave / COLT;
    const int ct   = wave % COLT;
    const int m    = lane & 15;
    const int hi   = lane >> 4;
    const int rowBase = blockIdx.x * ROWS;

    v8f acc = {0.f, 0.f, 0.f, 0.f, 0.f, 0.f, 0.f, 0.f};

    for (int kk = 0; kk < K; kk += KC) {
        // cooperative stage: W chunk [KC x OUT]
        for (int i = tid * 4; i < KC * OUT; i += 256 * 4) {
            int r = i / OUT, c = i % OUT;
            *(float4*)&sW[i] = *(const float4*)&W[(kk + r) * OUT + c];
        }
        // cooperative stage: X chunk [ROWS x KC] (row-clamped for tail tile)
        for (int i = tid * 4; i < ROWS * KC; i += 256 * 4) {
            int r = i / KC, c = i % KC;
            int gr = rowBase + r;
            if (gr > n - 1) gr = n - 1;
            *(float4*)&sX[i] = *(const float4*)&X[(size_t)gr * K + kk + c];
        }
        __syncthreads();

        const float* xrow = &sX[(rt * 16 + m) * KC];
        #pragma unroll
        for (int k0 = 0; k0 < KC; k0 += 4) {
            v2f a, b;
            a.x = xrow[k0 + 2 * hi];
            a.y = xrow[k0 + 2 * hi + 1];
            b.x = sW[(k0 + 2 * hi) * OUT + ct * 16 + m];
            b.y = sW[(k0 + 2 * hi + 1) * OUT + ct * 16 + m];
            acc = __builtin_amdgcn_wmma_f32_16x16x4_f32(
                /*neg_a=*/false, a, /*neg_b=*/false, b,
                /*c_mod=*/(short)0, acc, /*reuse_a=*/false, /*reuse_b=*/false);
        }
        __syncthreads();
    }

    const int col = ct * 16 + m;
    #pragma unroll
    for (int r = 0; r < 8; ++r) {
        int row = rowBase + rt * 16 + r + 8 * hi;
        if (row < n) T[(size_t)row * OUT + col] = acc[r];
    }
}

// ------------------------- message passing ---------------------------------

// agg[i] = tmp[i] * dinv[i]^2   (the self-loop term), float4 per thread
template <int OUT>
__global__ void selfloop_init_kernel(const float* __restrict__ tmp,
                                     const float* __restrict__ dinv,
                                     float* __restrict__ agg, int n) {
    constexpr int CH = OUT / 4;
    long long i = (long long)blockIdx.x * blockDim.x + threadIdx.x;
    long long total = (long long)n * CH;
    if (i >= total) return;
    int node = (int)(i / CH);
    int c    = (int)(i % CH) * 4;
    float s  = dinv[node];
    s = s * s;
    float4 v = *(const float4*)&tmp[(size_t)node * OUT + c];
    float4 o = {v.x * s, v.y * s, v.z * s, v.w * s};
    *(float4*)&agg[(size_t)node * OUT + c] = o;
}

// agg[dst] += tmp[src] * dinv[src]*dinv[dst], one (edge, float4-chunk)/thread
template <int OUT>
__global__ void scatter_edges_kernel(const long long* __restrict__ src,
                                     const long long* __restrict__ dst,
                                     const float* __restrict__ dinv,
                                     const float* __restrict__ tmp,
                                     float* __restrict__ agg, int e) {
    constexpr int CH = OUT / 4;
    long long total = (long long)e * CH;
    long long stride = (long long)gridDim.x * blockDim.x;
    for (long long i = (long long)blockIdx.x * blockDim.x + threadIdx.x;
         i < total; i += stride) {
        int ed = (int)(i / CH);           // CH is a power of two -> shifts
        int c  = (int)(i % CH) * 4;
        int s  = (int)src[ed];
        int d  = (int)dst[ed];
        float norm = dinv[s] * dinv[d];
        float4 v = *(const float4*)&tmp[(size_t)s * OUT + c];
        float* out = &agg[(size_t)d * OUT + c];
        atomicAdd(out + 0, v.x * norm);
        atomicAdd(out + 1, v.y * norm);
        atomicAdd(out + 2, v.z * norm);
        atomicAdd(out + 3, v.w * norm);
    }
}

// h = tanh(agg + b), in place
template <int OUT>
__global__ void bias_tanh_kernel(float* __restrict__ agg,
                                 const float* __restrict__ b, int n) {
    long long i = (long long)blockIdx.x * blockDim.x + threadIdx.x;
    long long total = (long long)n * OUT;
    if (i >= total) return;
    int f = (int)(i % OUT);
    agg[i] = tanhf(agg[i] + b[f]);
}

// ------------------------- MLP head ----------------------------------------
// out[i] = tanh(h2[i,:]@Wf1 + bf1) @ Wf2 + bf2; Wf1 64x32, Wf2 32x1.
__global__ void __launch_bounds__(256) mlp_head_kernel(
    const float* __restrict__ h2, const float* __restrict__ Wf1,
    const float* __restrict__ bf1, const float* __restrict__ Wf2,
    const float* __restrict__ bf2, float* __restrict__ out, int n) {
    __shared__ float sW1[64 * 32];
    __shared__ float sb1[32];
    __shared__ float sW2[32];

    const int tid = threadIdx.x;
    for (int i = tid * 4; i < 64 * 32; i += 256 * 4)
        *(float4*)&sW1[i] = *(const float4*)&Wf1[i];
    if (tid < 32) { sb1[tid] = bf1[tid]; sW2[tid] = Wf2[tid]; }
    __syncthreads();

    const int i = blockIdx.x * blockDim.x + tid;
    if (i >= n) return;

    float h[64];
    #pragma unroll
    for (int k = 0; k < 64; k += 4)
        *(float4*)&h[k] = *(const float4*)&h2[(size_t)i * 64 + k];

    float o = bf2[0];
    #pragma unroll
    for (int j = 0; j < 32; ++j) {
        float s = sb1[j];
        #pragma unroll
        for (int k = 0; k < 64; ++k) s = fmaf(h[k], sW1[k * 32 + j], s);
        o = fmaf(tanhf(s), sW2[j], o);
    }
    out[i] = o;
}

// ------------------------- launcher ----------------------------------------

extern "C" void kernel_launch(void* const* d_in, const int* in_sizes, int n_in,
                              void* d_out, int out_size, void* d_ws, size_t ws_size,
                              hipStream_t stream) {
    const float*     x   = (const float*)d_in[0];
    const long long* ei  = (const long long*)d_in[1];  // [2,E] int64
    const float*     W1  = (const float*)d_in[2];
    const float*     b1  = (const float*)d_in[3];
    const float*     W2  = (const float*)d_in[4];
    const float*     b2  = (const float*)d_in[5];
    const float*     Wf1 = (const float*)d_in[6];
    const float*     bf1 = (const float*)d_in[7];
    const float*     Wf2 = (const float*)d_in[8];
    const float*     bf2 = (const float*)d_in[9];
    float*           out = (float*)d_out;

    const int N = in_sizes[0] / 128;
    const int E = in_sizes[1] / 2;
    const long long* src = ei;
    const long long* dst = ei + E;

    // workspace layout (floats): dinv[N] | bufA[N*128] | bufB[N*128]
    float* dinv = (float*)d_ws;
    float* bufA = dinv + N;                       // tmp1, later tmp2+agg2
    float* bufB = bufA + (size_t)N * 128;         // agg1 -> h1
    float* tmp2 = bufA;
    float* agg2 = bufA + (size_t)N * 64;

    const int B = 256;
    // --- normalization ---
    deg_init_kernel<<<(N + B - 1) / B, B, 0, stream>>>(dinv, N);
    deg_edges_kernel<<<(E + B - 1) / B, B, 0, stream>>>(dst, dinv, E);
    dinv_kernel<<<(N + B - 1) / B, B, 0, stream>>>(dinv, N);

    // --- layer 1: 128 -> 128 ---
    gcn_transform_wmma<128><<<(N + 15) / 16, 256, 0, stream>>>(x, W1, bufA, N);
    {
        long long tot = (long long)N * 32;
        selfloop_init_kernel<128><<<(int)((tot + B - 1) / B), B, 0, stream>>>(bufA, dinv, bufB, N);
        scatter_edges_kernel<128><<<4096, B, 0, stream>>>(src, dst, dinv, bufA, bufB, E);
        long long el = (long long)N * 128;
        bias_tanh_kernel<128><<<(int)((el + B - 1) / B), B, 0, stream>>>(bufB, b1, N);
    }

    // --- layer 2: 128 -> 64 ---
    gcn_transform_wmma<64><<<(N + 31) / 32, 256, 0, stream>>>(bufB, W2, tmp2, N);
    {
        long long tot = (long long)N * 16;
        selfloop_init_kernel<64><<<(int)((tot + B - 1) / B), B, 0, stream>>>(tmp2, dinv, agg2, N);
        scatter_edges_kernel<64><<<4096, B, 0, stream>>>(src, dst, dinv, tmp2, agg2, E);
        long long el = (long long)N * 64;
        bias_tanh_kernel<64><<<(int)((el + B - 1) / B), B, 0, stream>>>(agg2, b2, N);
    }

    // --- MLP head ---
    mlp_head_kernel<<<(N + B - 1) / B, B, 0, stream>>>(agg2, Wf1, bf1, Wf2, bf2, out, N);
}